// GATLayer_25546465476890
// MI455X (gfx1250) — compile-verified
//
#include <hip/hip_runtime.h>

// ---------------------------------------------------------------------------
// GAT layer for MI455X (gfx1250, wave32).
// GEMM h = x@W + bias via V_WMMA_F32_16X16X4_F32 (exact fp32 math).
// Edge softmax/aggregation via L2-resident global float atomics.
// ---------------------------------------------------------------------------

typedef float v2f __attribute__((ext_vector_type(2)));
typedef float v8f __attribute__((ext_vector_type(8)));

#define N_NODES 50000
#define E_EDGES 800000
#define IN_F    128
#define OUT_F   64
#define NEG_SLOPE 0.2f
#define N_TILES (N_NODES / 16)   // 3125, exact

// Monotonic float -> uint key so atomicMax(u32) orders like float compare.
__device__ __forceinline__ unsigned fkey(float f) {
  unsigned u = __float_as_uint(f);
  return (u & 0x80000000u) ? ~u : (u | 0x80000000u);
}
__device__ __forceinline__ float unfkey(unsigned k) {
  return (k & 0x80000000u) ? __uint_as_float(k ^ 0x80000000u)
                           : __uint_as_float(~k);
}

// ---------------------------------------------------------------------------
// Kernel 0: zero the accumulated output, reset segment-max keys and denoms.
// (key 0 is below every real float's key -> behaves as -inf sentinel)
// ---------------------------------------------------------------------------
__global__ void gat_init(float* __restrict__ out,
                         unsigned* __restrict__ emaxkey,
                         float* __restrict__ denom) {
  int i = blockIdx.x * blockDim.x + threadIdx.x;
  if (i < N_NODES * OUT_F) out[i] = 0.0f;
  if (i < N_NODES) { emaxkey[i] = 0u; denom[i] = 0.0f; }
}

// ---------------------------------------------------------------------------
// Kernel 1: h = x@W + bias  (WMMA f32 16x16x4), fused Wh1/Wh2 row reductions.
// One wave -> one 16-row tile, 4 column tiles of 16 (OUT=64), K=128 in 4-steps.
//   A frag (16x4 f32): lanes 0-15 hold row M=lane, K={0,1}; lanes 16-31 K={2,3}
//   B frag (4x16 f32): lane holds col N=lane&15, K={0,1} (half 0) / {2,3} (half 1)
//   C/D (16x16 f32):   VGPR r: lanes 0-15 -> M=r, lanes 16-31 -> M=r+8; N=lane&15
// ---------------------------------------------------------------------------
__global__ void gat_gemm_wmma(const float* __restrict__ x,
                              const float* __restrict__ W,
                              const float* __restrict__ avec,
                              const float* __restrict__ bias,
                              float* __restrict__ h,
                              float* __restrict__ Wh1,
                              float* __restrict__ Wh2) {
  const int wave = threadIdx.x >> 5;
  const int tile = blockIdx.x * 4 + wave;
  if (tile >= N_TILES) return;          // wave-uniform: EXEC stays all-1s
  const int lane = threadIdx.x & 31;
  const int half = lane >> 4;           // 0 or 1
  const int lrow = lane & 15;
  const int m0 = tile * 16;

  v8f acc0 = {}, acc1 = {}, acc2 = {}, acc3 = {};
  const float* xrow = x + (size_t)(m0 + lrow) * IN_F + 2 * half;

  #pragma unroll 4
  for (int k = 0; k < IN_F; k += 4) {
    v2f a = *(const v2f*)(xrow + k);                       // x[m][k+2h .. +1]
    const float* wk = W + (size_t)(k + 2 * half) * OUT_F + lrow;
    v2f b0 = { wk[0],  wk[OUT_F +  0] };
    v2f b1 = { wk[16], wk[OUT_F + 16] };
    v2f b2 = { wk[32], wk[OUT_F + 32] };
    v2f b3 = { wk[48], wk[OUT_F + 48] };
    acc0 = __builtin_amdgcn_wmma_f32_16x16x4_f32(false, a, false, b0, (short)0, acc0, false, false);
    acc1 = __builtin_amdgcn_wmma_f32_16x16x4_f32(false, a, false, b1, (short)0, acc1, false, false);
    acc2 = __builtin_amdgcn_wmma_f32_16x16x4_f32(false, a, false, b2, (short)0, acc2, false, false);
    acc3 = __builtin_amdgcn_wmma_f32_16x16x4_f32(false, a, false, b3, (short)0, acc3, false, false);
  }

  // bias for this lane's column in each tile, attention weights a1/a2
  const float bc0 = bias[lrow],      bc1 = bias[16 + lrow];
  const float bc2 = bias[32 + lrow], bc3 = bias[48 + lrow];
  const float a1c0 = avec[lrow],      a1c1 = avec[16 + lrow];
  const float a1c2 = avec[32 + lrow], a1c3 = avec[48 + lrow];
  const float a2c0 = avec[64 + lrow],      a2c1 = avec[64 + 16 + lrow];
  const float a2c2 = avec[64 + 32 + lrow], a2c3 = avec[64 + 48 + lrow];

  #pragma unroll
  for (int r = 0; r < 8; ++r) {
    const int row = m0 + r + 8 * half;
    float h0 = acc0[r] + bc0;
    float h1 = acc1[r] + bc1;
    float h2 = acc2[r] + bc2;
    float h3 = acc3[r] + bc3;
    float* hp = h + (size_t)row * OUT_F + lrow;
    hp[0]  = h0;  hp[16] = h1;  hp[32] = h2;  hp[48] = h3;

    // per-row dot with a1/a2, reduced across the 16 lanes of this half
    float s1 = h0 * a1c0 + h1 * a1c1 + h2 * a1c2 + h3 * a1c3;
    float s2 = h0 * a2c0 + h1 * a2c1 + h2 * a2c2 + h3 * a2c3;
    #pragma unroll
    for (int m = 1; m < 16; m <<= 1) {   // xor masks 1,2,4,8 stay in-half
      s1 += __shfl_xor(s1, m, 32);
      s2 += __shfl_xor(s2, m, 32);
    }
    if (lrow == 0) { Wh1[row] = s1; Wh2[row] = s2; }
  }
}

// ---------------------------------------------------------------------------
// Kernel 2: e = leaky_relu(Wh1[src] + Wh2[dst]); segment max via key atomics.
// ---------------------------------------------------------------------------
__global__ void gat_edge_score(const float* __restrict__ Wh1,
                               const float* __restrict__ Wh2,
                               const int* __restrict__ src,
                               const int* __restrict__ dst,
                               float* __restrict__ escore,
                               unsigned* __restrict__ emaxkey) {
  int e = blockIdx.x * blockDim.x + threadIdx.x;
  if (e >= E_EDGES) return;
  const int s = src[e], d = dst[e];
  float v = Wh1[s] + Wh2[d];
  v = (v > 0.0f) ? v : NEG_SLOPE * v;
  escore[e] = v;
  atomicMax(emaxkey + s, fkey(v));
}

// ---------------------------------------------------------------------------
// Kernel 3: ex = exp(e - emax[src]); segment sum into denom.
// ---------------------------------------------------------------------------
__global__ void gat_edge_exp(const int* __restrict__ src,
                             const unsigned* __restrict__ emaxkey,
                             float* __restrict__ escore,
                             float* __restrict__ denom) {
  int e = blockIdx.x * blockDim.x + threadIdx.x;
  if (e >= E_EDGES) return;
  const int s = src[e];
  const float m = unfkey(emaxkey[s]);
  const float ex = __expf(escore[e] - m);
  escore[e] = ex;
  atomicAdd(denom + s, ex);
}

// ---------------------------------------------------------------------------
// Kernel 4: out[src] += (ex/denom[src]) * h[dst]; one thread per (edge,chan).
// h/out/denom are L2-resident (192MB L2), so gathers + atomics stay on-chip.
// ---------------------------------------------------------------------------
__global__ void gat_edge_accum(const int* __restrict__ src,
                               const int* __restrict__ dst,
                               const float* __restrict__ escore,
                               const float* __restrict__ denom,
                               const float* __restrict__ h,
                               float* __restrict__ out) {
  long long idx = (long long)blockIdx.x * blockDim.x + threadIdx.x;
  if (idx >= (long long)E_EDGES * OUT_F) return;
  const int e = (int)(idx >> 6);
  const int c = (int)(idx & 63);
  const int s = src[e], d = dst[e];
  const float alpha = escore[e] / denom[s];
  atomicAdd(out + (size_t)s * OUT_F + c, alpha * h[(size_t)d * OUT_F + c]);
}

// ---------------------------------------------------------------------------
extern "C" void kernel_launch(void* const* d_in, const int* in_sizes, int n_in,
                              void* d_out, int out_size, void* d_ws, size_t ws_size,
                              hipStream_t stream) {
  const float* x    = (const float*)d_in[0];   // [N, 128]
  const float* W    = (const float*)d_in[1];   // [128, 64]
  const float* avec = (const float*)d_in[2];   // [128, 1]
  const float* bias = (const float*)d_in[3];   // [64]
  const int*   src  = (const int*)d_in[4];     // [E]
  const int*   dst  = (const int*)d_in[5];     // [E]
  float* out = (float*)d_out;                  // [N, 64]

  // workspace layout (16.8 MB)
  float*    h        = (float*)d_ws;                       // N*64
  float*    Wh1      = h + (size_t)N_NODES * OUT_F;        // N
  float*    Wh2      = Wh1 + N_NODES;                      // N
  float*    denom    = Wh2 + N_NODES;                      // N
  unsigned* emaxkey  = (unsigned*)(denom + N_NODES);       // N
  float*    escore   = (float*)(emaxkey + N_NODES);        // E

  gat_init<<<(N_NODES * OUT_F + 255) / 256, 256, 0, stream>>>(out, emaxkey, denom);

  gat_gemm_wmma<<<(N_TILES + 3) / 4, 128, 0, stream>>>(x, W, avec, bias, h, Wh1, Wh2);

  gat_edge_score<<<(E_EDGES + 255) / 256, 256, 0, stream>>>(Wh1, Wh2, src, dst,
                                                            escore, emaxkey);
  gat_edge_exp<<<(E_EDGES + 255) / 256, 256, 0, stream>>>(src, emaxkey, escore, denom);

  const long long tot = (long long)E_EDGES * OUT_F;
  gat_edge_accum<<<(unsigned)((tot + 255) / 256), 256, 0, stream>>>(src, dst, escore,
                                                                    denom, h, out);
}